// JitterBatch_65541200937284
// MI455X (gfx1250) — compile-verified
//
#include <hip/hip_runtime.h>

// Problem shape (from reference): x[64][3][512][512] fp32, per-batch shifts in [-32,32].
#define B_DIM 64
#define C_DIM 3
#define H_DIM 512
#define W_DIM 512
#define LIM   32

#define ROWS_PER_BLOCK 8
#define THREADS        256
#define CHUNKS_PER_BLOCK (ROWS_PER_BLOCK * (W_DIM / 4))   // 1024 float4 chunks
#define ITERS            (CHUNKS_PER_BLOCK / THREADS)     // 4

// CDNA5 async global->LDS DMA (ASYNCcnt pipeline); falls back to
// b128 load + ds_store staging if the toolchain lacks the builtin.
#if defined(__has_builtin)
#if __has_builtin(__builtin_amdgcn_global_load_async_to_lds_b128) && \
    __has_builtin(__builtin_amdgcn_s_wait_asynccnt)
#define USE_ASYNC_LDS 1
#endif
#endif

typedef float float4v __attribute__((ext_vector_type(4)));

#ifdef USE_ASYNC_LDS
// Builtin signature (verified via hipcc diagnostic): param0 is a pointer to
// '__attribute__((vector_size(16))) int' in addrspace(1) (global), param1 the
// same in addrspace(3) (LDS), then two integer immediates (offset, cpol).
typedef int v4i_t __attribute__((vector_size(16)));
typedef __attribute__((address_space(1))) v4i_t* g_v4i_ptr;
typedef __attribute__((address_space(3))) v4i_t* l_v4i_ptr;
#endif

__global__ __launch_bounds__(THREADS) void jitter_roll_kernel(
    const float* __restrict__ x,
    const int* __restrict__ off1,
    const int* __restrict__ off2,
    float* __restrict__ out)
{
    __shared__ float smem[ROWS_PER_BLOCK * W_DIM];   // 16 KB of the 320 KB WGP LDS

    const int blocksPerImage = H_DIM / ROWS_PER_BLOCK;     // 64
    const int bc = blockIdx.x / blocksPerImage;            // fused (b, c)
    const int h0 = (blockIdx.x % blocksPerImage) * ROWS_PER_BLOCK;
    const int b  = bc / C_DIM;

    // Per-batch shifts (uniform across block -> scalarized by compiler).
    const int s1 = off1[b] - LIM;   // shift along W (last dim)
    const int s2 = off2[b] - LIM;   // shift along H

    const float* imgIn  = x   + (size_t)bc * H_DIM * W_DIM;
    float*       imgOut = out + (size_t)bc * H_DIM * W_DIM;

    const int tid = threadIdx.x;

    // ---- Phase 1: stage 8 H-rolled input rows into LDS via async DMA,
    //      row-aligned b128, no VGPR staging. ----
#pragma unroll
    for (int it = 0; it < ITERS; ++it) {
        const int i   = tid + it * THREADS;     // chunk id 0..1023
        const int row = i >> 7;                 // 128 float4 chunks per row
        const int c16 = i & 127;
        const int hin = (h0 + row - s2) & (H_DIM - 1);   // roll in H, two's-complement safe
        const float* src = imgIn + (size_t)hin * W_DIM + c16 * 4;
        float*       dst = &smem[row * W_DIM + c16 * 4];
#ifdef USE_ASYNC_LDS
        __builtin_amdgcn_global_load_async_to_lds_b128(
            (g_v4i_ptr)src, (l_v4i_ptr)dst,
            /*imm offset*/0, /*cpol*/0);
#else
        *(float4v*)dst = *(const float4v*)src;
#endif
    }
#ifdef USE_ASYNC_LDS
    __builtin_amdgcn_s_wait_asynccnt(0);
#endif
    __syncthreads();

    // ---- Phase 2: W-rotation resolved in LDS; aligned non-temporal b128
    //      stores (single-pass stream: keep the output out of the caches). ----
#pragma unroll
    for (int it = 0; it < ITERS; ++it) {
        const int i     = tid + it * THREADS;
        const int row   = i >> 7;
        const int c16   = i & 127;
        const int wbase = c16 * 4;
        float4v v;
#pragma unroll
        for (int j = 0; j < 4; ++j) {
            const int win = (wbase + j - s1) & (W_DIM - 1);   // roll in W
            v[j] = smem[row * W_DIM + win];
        }
        float4v* dst = (float4v*)(imgOut + (size_t)(h0 + row) * W_DIM + wbase);
        __builtin_nontemporal_store(v, dst);
    }
}

extern "C" void kernel_launch(void* const* d_in, const int* in_sizes, int n_in,
                              void* d_out, int out_size, void* d_ws, size_t ws_size,
                              hipStream_t stream)
{
    const float* x    = (const float*)d_in[0];
    const int*   off1 = (const int*)d_in[1];
    const int*   off2 = (const int*)d_in[2];
    float*       out  = (float*)d_out;

    const int blocks = B_DIM * C_DIM * (H_DIM / ROWS_PER_BLOCK);   // 12288
    jitter_roll_kernel<<<blocks, THREADS, 0, stream>>>(x, off1, off2, out);
}